// SelfAttention_Text_77275051590042
// MI455X (gfx1250) — compile-verified
//
#include <hip/hip_runtime.h>

typedef float v2f __attribute__((ext_vector_type(2)));
typedef float v8f __attribute__((ext_vector_type(8)));

#define CDIM   1024
#define BROWS  4
#define WAVES  16                 // waves per workgroup (wave32 -> 512 threads)
#define KCHUNK (CDIM / WAVES)     // 64 K per wave, 16 WMMA steps

// Computes out[m, n0:n0+16] = sum_k A[m,k]*W[k,n] + bias[n] for m = 0..3.
// A is [4,1024] row-major, W is [1024,1024] row-major ([in,out]), bias [1024].
// One workgroup per 16-column N tile; 16 waves split the K dimension; partial
// 4x16 tiles are reduced deterministically through LDS.
__global__ __launch_bounds__(WAVES * 32)
void gemm4xC_wmma_f32(const float* __restrict__ A,
                      const float* __restrict__ W,
                      const float* __restrict__ bias,
                      float* __restrict__ out) {
    __shared__ float red[WAVES * BROWS * 16];

    const int tid   = threadIdx.x;
    const int lane  = tid & 31;
    const int wave  = tid >> 5;
    const int n0    = blockIdx.x * 16;
    const int nlane = lane & 15;          // N (for B/C) or M (for A) within tile
    const int khalf = (lane >> 4) * 2;    // lanes 16-31 handle K-half {2,3}

    // A-fragment row for this lane: M = nlane; rows 4..15 are don't-care, so
    // clamp to a valid row (never reads out of bounds; result rows unused).
    const float* Arow = A + (nlane & (BROWS - 1)) * CDIM;

    v8f c = {};
    const int kbase = wave * KCHUNK;

#pragma unroll
    for (int kk = 0; kk < KCHUNK; kk += 4) {
        const int k = kbase + kk + khalf;
        v2f a, b;
        a.x = Arow[k];
        a.y = Arow[k + 1];
        b.x = W[k * CDIM + n0 + nlane];
        b.y = W[(k + 1) * CDIM + n0 + nlane];
        // D = A(16x4) * B(4x16) + C ; f32 everywhere.
        c = __builtin_amdgcn_wmma_f32_16x16x4_f32(
                /*neg_a=*/false, a, /*neg_b=*/false, b,
                /*c_mod=*/(short)0, c, /*reuse_a=*/false, /*reuse_b=*/false);
    }

    // C/D layout: VGPR r, lanes 0-15 hold row M=r, N=lane. Stash rows 0..3.
    if (lane < 16) {
#pragma unroll
        for (int r = 0; r < BROWS; ++r)
            red[wave * (BROWS * 16) + r * 16 + lane] = c[r];
    }
    __syncthreads();

    // Deterministic fixed-order reduction over the 16 K-split partials.
    if (tid < BROWS * 16) {
        const int r = tid >> 4;
        const int n = tid & 15;
        float s = 0.0f;
#pragma unroll
        for (int w = 0; w < WAVES; ++w)
            s += red[w * (BROWS * 16) + r * 16 + n];
        out[r * CDIM + n0 + n] = s + bias[n0 + n];
    }
}

// Reference collapses exactly to: out = (x @ Wv + bv) @ Wo + bo   (shape [4,1024])
// (broadcast tokens -> uniform softmax -> attention output == v; Wq/Wk unused)
extern "C" void kernel_launch(void* const* d_in, const int* in_sizes, int n_in,
                              void* d_out, int out_size, void* d_ws, size_t ws_size,
                              hipStream_t stream) {
    (void)in_sizes; (void)n_in; (void)out_size; (void)ws_size;
    const float* x  = (const float*)d_in[0];
    const float* Wv = (const float*)d_in[5];
    const float* bv = (const float*)d_in[6];
    const float* Wo = (const float*)d_in[7];
    const float* bo = (const float*)d_in[8];
    float* v   = (float*)d_ws;     // [4,1024] intermediate, 16 KB
    float* out = (float*)d_out;    // [4,1024]

    dim3 grid(CDIM / 16);          // 64 N-tiles
    dim3 block(WAVES * 32);        // 512 threads = 16 wave32

    gemm4xC_wmma_f32<<<grid, block, 0, stream>>>(x, Wv, bv, v);
    gemm4xC_wmma_f32<<<grid, block, 0, stream>>>(v, Wo, bo, out);
}